// GRU_Conv_38431367365140
// MI455X (gfx1250) — compile-verified
//
#include <hip/hip_runtime.h>
#include <math.h>

typedef __attribute__((ext_vector_type(2))) float v2f;
typedef __attribute__((ext_vector_type(8))) float v8f;

#define TAPS 261
#define HP   272      // padded hidden (17*16)
#define NT   17       // N tiles of 16
#define KK   68       // K-steps of 4 (272/4)
#define BQ   16       // batch
#define LSEQ 2000
#define CT   130      // (TAPS-1)/2
#define NOUT 1740     // LSEQ - TAPS + 1
#define NTHREADS (NT*32)

struct KParams {
  const float* x;            // (16,2000,2) complex64 as float pairs
  const float* wi[2][3];     // W_ir, W_iz, W_in  per gru  (2 x 261 complex)
  const float* bias[2][4];   // b_ir, b_iz, b_in, b_hn     (261 complex)
  const float4* wfrag;       // pre-swizzled W_h fragments in workspace
  float2* out;               // (16,1740,2) complex64
};
struct PParams {
  const float* wh[2][3];     // W_hr, W_hz, W_hn per gru (261 x 261 complex)
  float4* wfrag;
};

__device__ __forceinline__ v8f wmma4(v2f a, v2f b, v8f c){
  // D = A(16x4) * B(4x16) + C, fp32 WMMA
  return __builtin_amdgcn_wmma_f32_16x16x4_f32(false, a, false, b, (short)0, c,
                                               false, false);
}
__device__ __forceinline__ float2 cmul(float2 a, float2 b){
  return make_float2(a.x*b.x - a.y*b.y, a.x*b.y + a.y*b.x);
}
__device__ __forceinline__ float2 cadd(float2 a, float2 b){
  return make_float2(a.x + b.x, a.y + b.y);
}
__device__ __forceinline__ float sigm(float v){ return 1.0f/(1.0f + __expf(-v)); }

// ---------------------------------------------------------------------------
// Swizzle recurrence weights (261x261 complex, row=k input, col=n output) into
// per-lane WMMA B fragments: wfrag[((e*3+g)*NT+tile)*KK+kk][lane] = float4
// (Wre[k0], Wre[k0+1], Wim[k0], Wim[k0+1]) with n = tile*16+(lane&15),
// k0 = kk*4 + 2*(lane>>4).  Zero-padded beyond 261.
// ---------------------------------------------------------------------------
__global__ void prep_w(PParams P){
  int i = blockIdx.x * blockDim.x + threadIdx.x;
  const int TOT = 2*3*NT*KK*32;
  if (i >= TOT) return;
  int lane = i & 31;
  int s = i >> 5;
  int kk   = s % KK;  s /= KK;
  int tile = s % NT;  s /= NT;
  int g    = s % 3;   s /= 3;
  int e    = s;
  int n  = tile*16 + (lane & 15);
  int k0 = kk*4 + 2*(lane >> 4);
  const float* W = P.wh[e][g];
  float4 v = make_float4(0.f, 0.f, 0.f, 0.f);
  if (n < TAPS){
    if (k0     < TAPS){ v.x = W[(k0*TAPS+n)*2];     v.z = W[(k0*TAPS+n)*2+1]; }
    if (k0 + 1 < TAPS){ v.y = W[((k0+1)*TAPS+n)*2]; v.w = W[((k0+1)*TAPS+n)*2+1]; }
  }
  P.wfrag[i] = v;
}

// ---------------------------------------------------------------------------
// One block per GRU (e = 0:x, 1:y).  17 waves, wave w owns output tile w.
// h kept in LDS in A-fragment layout; recurrence via V_WMMA_F32_16X16X4_F32
// using 3-mult Karatsuba complex GEMM; sliding-window contraction fused
// (h_t is exactly the tap vector of output column t-130).
// ---------------------------------------------------------------------------
__global__ __launch_bounds__(NTHREADS) void gru_main(KParams P){
  __shared__ float4 s_afrag[KK*32];       // h fragments  (hr0,hr1,hi0,hi1)
  __shared__ float2 s_wi[3][2][HP];       // input weights per gate
  __shared__ float2 s_b[4][HP];           // b_ir, b_iz, b_in, b_hn
  __shared__ float2 s_xt[2][BQ][2];       // double-buffered x_t
  __shared__ float2 s_accE[BQ];           // streamed output accumulator

  const int e    = blockIdx.x;
  const int tid  = threadIdx.x;
  const int wv   = tid >> 5;
  const int lane = tid & 31;
  const int tile = wv;
  const int kh   = lane >> 4;
  const int nn   = lane & 15;
  const int hcol = tile*16 + nn;

  // ---- one-time LDS setup (h0 = 0, small weights, biases) ----
  for (int i = tid; i < KK*32; i += NTHREADS) s_afrag[i] = make_float4(0,0,0,0);
  for (int i = tid; i < 3*2*HP; i += NTHREADS){
    int g = i / (2*HP); int r = i % (2*HP); int k = r / HP; int c = r % HP;
    float2 v = make_float2(0,0);
    if (c < TAPS) v = ((const float2*)P.wi[e][g])[k*TAPS + c];
    s_wi[g][k][c] = v;
  }
  for (int i = tid; i < 4*HP; i += NTHREADS){
    int b = i / HP; int c = i % HP;
    float2 v = make_float2(0,0);
    if (c < TAPS) v = ((const float2*)P.bias[e][b])[c];
    s_b[b][c] = v;
  }
  if (tid < BQ) s_accE[tid] = make_float2(0,0);
  __syncthreads();

  const float2* x2 = (const float2*)P.x;
  const float4* wf = P.wfrag + (size_t)(e*3) * NT * KK * 32;

  for (int t = 0; t < LSEQ; ++t){
    // stage x_t (double buffered; read after B1 this iteration)
    if (wv == 0){
      int m = lane >> 1, c = lane & 1;
      s_xt[t & 1][m][c] = x2[(m*LSEQ + t)*2 + c];
    }
    __syncthreads();                               // B1: h writes + xt visible

    // flush previous step's streamed output (disjoint from wmma work below)
    {
      int tp = t - 1;
      if (wv == 1 && lane < BQ && tp >= CT && tp < CT + NOUT){
        int tt = tp - CT;
        P.out[(lane*NOUT + tt)*2 + e] = s_accE[lane];
        s_accE[lane] = make_float2(0,0);
      }
    }

    // ---- recurrence matmuls: acc[g] = h @ W_h[g]  (complex, fp32 WMMA) ----
    // Karatsuba: P1 = hr*Wr, P2 = hi*Wi, P3 = (hr+hi)*(Wr+Wi)
    //            Re = P1 - P2, Im = P3 - P1 - P2  (fixup once per step)
    v8f zero = {};
    v8f p1[3], p2[3], p3[3];
    #pragma unroll
    for (int g = 0; g < 3; ++g){ p1[g] = zero; p2[g] = zero; p3[g] = zero; }

    for (int kk = 0; kk < KK; ++kk){
      float4 av = s_afrag[kk*32 + lane];           // ds_load_b128
      v2f ar = { av.x,        av.y };
      v2f ai = { av.z,        av.w };
      v2f as = { av.x + av.z, av.y + av.w };
      #pragma unroll
      for (int g = 0; g < 3; ++g){
        float4 bv = wf[(((size_t)g*NT + tile)*KK + kk)*32 + lane]; // L2 hit
        v2f br = { bv.x,        bv.y };
        v2f bi = { bv.z,        bv.w };
        v2f bs = { bv.x + bv.z, bv.y + bv.w };
        p1[g] = wmma4(ar, br, p1[g]);
        p2[g] = wmma4(ai, bi, p2[g]);
        p3[g] = wmma4(as, bs, p3[g]);
      }
    }
    // per-step complex fixup on full 16x16 fragments
    v8f gR[3], gI[3];
    #pragma unroll
    for (int g = 0; g < 3; ++g){
      gR[g] = p1[g] - p2[g];
      gI[g] = p3[g] - p1[g] - p2[g];
    }
    __syncthreads();                               // B2: all h reads done

    // ---- gates, state update, fused sliding-window contraction ----
    const bool doE = (t >= CT) && (t < CT + NOUT);
    const int  tt  = t - CT;
    #pragma unroll
    for (int j = 0; j < 8; ++j){
      int m = j + 8*kh;                            // C/D layout: M = j + 8*(lane/16)
      float2 xa = s_xt[t & 1][m][0];
      float2 xb = s_xt[t & 1][m][1];

      float2 rin = cadd(cadd(cmul(xa, s_wi[0][0][hcol]), cmul(xb, s_wi[0][1][hcol])),
                        s_b[0][hcol]);
      rin.x += gR[0][j]; rin.y += gI[0][j];
      float2 zin = cadd(cadd(cmul(xa, s_wi[1][0][hcol]), cmul(xb, s_wi[1][1][hcol])),
                        s_b[1][hcol]);
      zin.x += gR[1][j]; zin.y += gI[1][j];
      float2 hn = make_float2(s_b[3][hcol].x + gR[2][j], s_b[3][hcol].y + gI[2][j]);

      float2 rr = make_float2(sigm(rin.x), sigm(rin.y));
      float2 zz = make_float2(sigm(zin.x), sigm(zin.y));
      float2 nin = cadd(cadd(cmul(xa, s_wi[2][0][hcol]), cmul(xb, s_wi[2][1][hcol])),
                        s_b[2][hcol]);
      nin = cadd(nin, cmul(rr, hn));
      float2 nv = make_float2(tanhf(nin.x), tanhf(nin.y));

      // h_new = (1-z)*n + z*h_old
      int kkh = hcol >> 2, khh = (hcol >> 1) & 1, fe = hcol & 1;
      float* af = (float*)&s_afrag[kkh*32 + m + 16*khh];
      float2 hold = make_float2(af[fe], af[2+fe]);
      float2 omz  = make_float2(1.f - zz.x, -zz.y);
      float2 hnew = cadd(cmul(omz, nv), cmul(zz, hold));

      float2 p = make_float2(0,0);
      if (hcol < TAPS){
        af[fe]   = hnew.x;                         // write back own columns only
        af[2+fe] = hnew.y;
        if (doE){
          float2 xe = x2[(m*LSEQ + tt + hcol)*2 + e];
          p = cmul(hnew, xe);                      // tap contribution
        }
      }
      // reduce over the 16 taps of this tile (lanes sharing kh-half)
      for (int msk = 1; msk < 16; msk <<= 1){
        p.x += __shfl_xor(p.x, msk, 32);
        p.y += __shfl_xor(p.y, msk, 32);
      }
      if (doE && nn == 0){
        atomicAdd(&s_accE[m].x, p.x);              // ds_add_f32, 17 tiles
        atomicAdd(&s_accE[m].y, p.y);
      }
    }
  }
}

extern "C" void kernel_launch(void* const* d_in, const int* in_sizes, int n_in,
                              void* d_out, int out_size, void* d_ws, size_t ws_size,
                              hipStream_t stream){
  (void)in_sizes; (void)n_in; (void)out_size; (void)ws_size;
  // d_in: [0]=x, [1..10]=params_x leaves, [11..20]=params_y leaves in dict order
  // (W_ir W_iz W_in W_hr W_hz W_hn b_ir b_iz b_in b_hn)
  KParams kp;
  PParams pp;
  kp.x = (const float*)d_in[0];
  for (int e = 0; e < 2; ++e){
    int base = 1 + 10*e;
    for (int g = 0; g < 3; ++g){
      kp.wi[e][g] = (const float*)d_in[base + g];       // W_i{r,z,n}
      pp.wh[e][g] = (const float*)d_in[base + 3 + g];   // W_h{r,z,n}
    }
    for (int b = 0; b < 4; ++b)
      kp.bias[e][b] = (const float*)d_in[base + 6 + b]; // b_ir b_iz b_in b_hn
  }
  pp.wfrag = (float4*)d_ws;
  kp.wfrag = (const float4*)d_ws;
  kp.out   = (float2*)d_out;

  const int TOT = 2*3*NT*KK*32;
  prep_w<<<(TOT + 255)/256, 256, 0, stream>>>(pp);
  gru_main<<<2, NTHREADS, 0, stream>>>(kp);
}